// DeltaQuantLinear_87540023427416
// MI455X (gfx1250) — compile-verified
//
#include <hip/hip_runtime.h>

// DeltaQuantLinear for MI455X (gfx1250, wave32, WMMA + Tensor Data Mover).
//
// y = x @ (W + (Q - zp)*scale)^T + bias
//   = x@W^T + scale[o]*( x@Q^T - zp[o]*sum_x[t] ) + bias[o]
//
// Memory-bound: 360 MB (W f32 + Q i32) streamed once -> ~15.5 us floor at
// 23.3 TB/s. Saturating HBM needs ~14 MB in flight (BW x ~600ns latency);
// VGPR-bound b64 loads can't get there with 688 waves, so weights/quants are
// streamed by the TDM into LDS, 4 stages deep (24 KB in flight per wave,
// ~16 MB chip-wide). Compute: V_WMMA_F32_16X16X4_F32, two accumulators
// sharing one A operand, exact f32 math.

typedef float v2f  __attribute__((ext_vector_type(2)));
typedef float v8f  __attribute__((ext_vector_type(8)));
typedef int   v2i  __attribute__((ext_vector_type(2)));
typedef unsigned int u32x4 __attribute__((ext_vector_type(4)));
typedef int   i32x4 __attribute__((ext_vector_type(4)));
typedef int   i32x8 __attribute__((ext_vector_type(8)));

static constexpr int IN_F   = 4096;
static constexpr int OUT_F  = 11008;
static constexpr int TOKENS = 8;

static constexpr int CHUNK  = 64;            // K floats per pipeline stage
static constexpr int ROWF   = CHUNK + 1;     // LDS row pitch (TDM pad: 64 DW + 1 DW)
static constexpr int TILEF  = 16 * ROWF;     // 1040 floats per 16-row tile
static constexpr int STAGES = 4;             // pipeline depth
static constexpr int WAVES  = 4;             // waves per block
static constexpr int NCHUNK = IN_F / CHUNK;  // 64
static constexpr int WAVE_F = STAGES * 2 * TILEF;          // floats per wave
static constexpr int SMEM_BYTES = WAVES * WAVE_F * 4;      // 133,120 B

// ---------------------------------------------------------------------------
// TDM: 2D tile load, 16 rows x CHUNK f32 elements, row pitch 4096 elements,
// +1 DWORD LDS padding per 64 DWORDs (=> 65-float LDS row pitch).
// D# layout per CDNA5 ISA ch.8 (groups 0/1; groups 2/3 zero for 2D).
// ---------------------------------------------------------------------------
__device__ __forceinline__ void tdm_load_tile(unsigned int lds_byte,
                                              const void* gptr) {
    unsigned long long ga = (unsigned long long)(uintptr_t)gptr;

    u32x4 g0;
    g0.x = 1u;                                   // count=1 (valid), user mode
    g0.y = lds_byte;                             // lds_addr [63:32]
    g0.z = (unsigned int)ga;                     // global_addr [96:64]
    g0.w = (unsigned int)((ga >> 32) & 0x1FFFFFFu)  // global_addr [120:96]
         | (2u << 30);                           // type=2 ("image")

    i32x8 g1;
    g1[0] = (2 << 16)                            // data_size = 4 bytes
          | (1 << 20)                            // pad_enable
          | (5 << 22);                           // pad_interval: 64 DWORDs
                                                 // pad_amount bits[31:25]=0 -> 1 DW
    g1[1] = (int)((unsigned)CHUNK << 16);        // tensor_dim0 = 64 (== tile, no OOB)
    g1[2] = (int)(16u << 16);                    // tensor_dim1 = 16
    g1[3] = (int)((unsigned)CHUNK << 16);        // tile_dim0 = 64
    g1[4] = 16;                                  // tile_dim1 = 16, tile_dim2 = 0
    g1[5] = IN_F;                                // tensor_dim0_stride = 4096 elems
    g1[6] = 0;
    g1[7] = 0;

    i32x4 z4 = {0, 0, 0, 0};
#if defined(__clang_major__) && (__clang_major__ >= 23)
    i32x8 z8 = {0, 0, 0, 0, 0, 0, 0, 0};
    __builtin_amdgcn_tensor_load_to_lds(g0, g1, z4, z4, z8, 0);
#else
    __builtin_amdgcn_tensor_load_to_lds(g0, g1, z4, z4, 0);
#endif
}

// ---------------------------------------------------------------------------
// Kernel 0: sum_x[t] = sum_i x[t,i]
// ---------------------------------------------------------------------------
__global__ __launch_bounds__(256)
void dql_sumx_kernel(const float* __restrict__ x, float* __restrict__ sums) {
    __shared__ float red[256];
    const int t   = blockIdx.x;
    const int tid = threadIdx.x;
    float s = 0.0f;
    for (int i = tid; i < IN_F; i += 256)
        s += x[t * IN_F + i];
    red[tid] = s;
    __syncthreads();
    for (int off = 128; off > 0; off >>= 1) {
        if (tid < off) red[tid] += red[tid + off];
        __syncthreads();
    }
    if (tid == 0) sums[t] = red[0];
}

// ---------------------------------------------------------------------------
// Kernel 1: TDM-pipelined dual-accumulator WMMA GEMM + fused dequant epilogue.
// One wave = one 16-column output tile. 4 waves/block, 172 blocks.
// ---------------------------------------------------------------------------
extern __shared__ float smem[];

__global__ __launch_bounds__(128)
void dql_wmma_tdm_kernel(const float* __restrict__ x,
                         const float* __restrict__ bw,
                         const int*   __restrict__ qd,
                         const float* __restrict__ scales,
                         const float* __restrict__ zps,
                         const float* __restrict__ bias,
                         const float* __restrict__ sumx,
                         float*       __restrict__ out) {
    const int lane = threadIdx.x & 31;
    // wave id: uniform per wave; readfirstlane so TDM descriptors are scalar.
    const int wave = __builtin_amdgcn_readfirstlane((int)(threadIdx.x >> 5));
    const int tile = blockIdx.x * WAVES + wave;   // 0..687
    const int n0   = tile * 16;

    const int nl   = lane & 15;                   // N within tile
    const int n    = n0 + nl;
    const int hh   = lane >> 4;                   // half-wave -> K pair select
    const int m    = lane & 15;                   // A row
    const int koff = 2 * hh;

    float* wv = smem + wave * WAVE_F;             // this wave's LDS region
    const unsigned int ldsbase =
        (unsigned int)(uintptr_t)wv;              // LDS byte address (flat[31:0])

    const float* __restrict__ xrow = x + (size_t)(m & 7) * IN_F;
    const size_t wrow_off = (size_t)n0 * IN_F;

    // ---- pipeline prologue: fill STAGES stages (2 tensor ops per stage) ----
    for (int s = 0; s < STAGES; ++s) {
        const int k0c = s * CHUNK;
        tdm_load_tile(ldsbase + (unsigned)(s * 2 + 0) * TILEF * 4,
                      bw + wrow_off + k0c);
        tdm_load_tile(ldsbase + (unsigned)(s * 2 + 1) * TILEF * 4,
                      qd + wrow_off + k0c);
    }

    v8f accB = {};
    v8f accQ = {};

    for (int c = 0; c < NCHUNK; ++c) {
        // Oldest chunk's 2 TDM ops complete (in-order per wave).
        __builtin_amdgcn_s_wait_tensorcnt((short)(2 * (STAGES - 1)));
        asm volatile("" ::: "memory");

        const int    st = c & (STAGES - 1);
        const float* wt = wv + (st * 2 + 0) * TILEF + nl * ROWF;
        const int*   qt = (const int*)(wv + (st * 2 + 1) * TILEF) + nl * ROWF;
        const float* xk = xrow + c * CHUNK;

#pragma unroll 4
        for (int ks = 0; ks < CHUNK; ks += 4) {
            const int kk = ks + koff;
            v2f a = *(const v2f*)(xk + kk);     // A frag: x (L2-resident)
            v2f b = *(const v2f*)(wt + kk);     // B frag: W tile from LDS
            v2i q = *(const v2i*)(qt + kk);     // B frag: Q tile from LDS
            v2f bq;
            bq.x = (float)q.x;
            bq.y = (float)q.y;

            accB = __builtin_amdgcn_wmma_f32_16x16x4_f32(
                false, a, false, b, (short)0, accB, false, false);
            accQ = __builtin_amdgcn_wmma_f32_16x16x4_f32(
                false, a, false, bq, (short)0, accQ, false, false);
        }

        if (c + STAGES < NCHUNK) {
            // WAR guard: LDS reads of this buffer retired before TDM rewrites it.
            asm volatile("s_wait_dscnt 0" ::: "memory");
            const int k0c = (c + STAGES) * CHUNK;
            tdm_load_tile(ldsbase + (unsigned)(st * 2 + 0) * TILEF * 4,
                          bw + wrow_off + k0c);
            tdm_load_tile(ldsbase + (unsigned)(st * 2 + 1) * TILEF * 4,
                          qd + wrow_off + k0c);
        }
    }

    // ---- epilogue: C layout acc[r] -> M = r + 8*hh, N = nl. Tokens in hh==0.
    const float sc = scales[n];
    const float z  = zps[n];
    const float bs = bias[n];
    if (hh == 0) {
#pragma unroll
        for (int r = 0; r < TOKENS; ++r) {
            const float sx  = sumx[r];
            out[(size_t)r * OUT_F + n] = accB[r] + sc * (accQ[r] - z * sx) + bs;
        }
    }
}

// ---------------------------------------------------------------------------
// Launch
// ---------------------------------------------------------------------------
extern "C" void kernel_launch(void* const* d_in, const int* in_sizes, int n_in,
                              void* d_out, int out_size, void* d_ws, size_t ws_size,
                              hipStream_t stream) {
    (void)in_sizes; (void)n_in; (void)out_size; (void)ws_size;

    const float* x      = (const float*)d_in[0];  // [8, 4096]
    const float* bw     = (const float*)d_in[1];  // [11008, 4096]
    const int*   qd     = (const int*)  d_in[2];  // [11008, 4096]
    const float* scales = (const float*)d_in[3];  // [11008]
    const float* zps    = (const float*)d_in[4];  // [11008]
    const float* bias   = (const float*)d_in[5];  // [11008]
    float*       out    = (float*)d_out;          // [8, 11008]
    float*       sums   = (float*)d_ws;           // 8 floats of scratch

    dql_sumx_kernel<<<TOKENS, 256, 0, stream>>>(x, sums);

    const int tiles  = OUT_F / 16;        // 688
    const int blocks = tiles / WAVES;     // 172
    dql_wmma_tdm_kernel<<<blocks, 128, SMEM_BYTES, stream>>>(
        x, bw, qd, scales, zps, bias, sums, out);
}